// FFGM_30777735643666
// MI455X (gfx1250) — compile-verified
//
#include <hip/hip_runtime.h>
#include <math.h>

// ---------------------------------------------------------------------------
// Problem constants (B=1, C=64, H=W=192, 7x7 deform kernel, 8x8 FFT patches)
// ---------------------------------------------------------------------------
#define HIMG 192
#define WIMG 192
#define P_TOT (HIMG * WIMG)      // 36864 pixels
#define CCH 64
#define KDEF 49
#define LDA 68                   // LDS A stride (68 % 64 == 4 -> conflict free)
#define LDA2 132                 // LDS A stride for K=128 (132 % 64 == 4)

typedef __attribute__((ext_vector_type(2))) float v2f;
typedef __attribute__((ext_vector_type(8))) float v8f;

// D = A(16x4 f32) * B(4x16 f32) + C(16x16 f32)   -- CDNA5 V_WMMA_F32_16X16X4_F32
__device__ __forceinline__ v8f wmma4(v2f a, v2f b, v8f c) {
  return __builtin_amdgcn_wmma_f32_16x16x4_f32(false, a, false, b, (short)0, c,
                                               false, false);
}

// A fragment: lanes 0-15 hold M=lane, K = kb+0/kb+1; lanes 16-31 hold K=kb+2/+3
__device__ __forceinline__ v2f afrag(const float* As, int lane, int kb, int lda) {
  const float* p = As + (lane & 15) * lda + kb + ((lane >> 4) << 1);
  v2f a;
  a.x = p[0];
  a.y = p[1];
  return a;
}

// B fragment from Bs[K][ldb]: lanes 0-15 N=lane rows kb,kb+1; lanes 16-31 rows kb+2,kb+3
__device__ __forceinline__ v2f bfrag(const float* Bs, int lane, int kb, int n0,
                                     int ldb) {
  int k = kb + ((lane >> 4) << 1);
  int n = n0 + (lane & 15);
  v2f b;
  b.x = Bs[k * ldb + n];
  b.y = Bs[(k + 1) * ldb + n];
  return b;
}

// ---------------------------------------------------------------------------
// 1) NCHW[64][P] -> HWC[P][64] LDS-tiled transpose (coalesced both sides)
// ---------------------------------------------------------------------------
__global__ __launch_bounds__(256) void k_transpose(const float* __restrict__ S,
                                                   float* __restrict__ D) {
  __shared__ float tile[64][65];
  int t = threadIdx.x;
  int p0 = blockIdx.x * 64;
#pragma unroll
  for (int i = 0; i < 16; ++i) {
    int li = t + i * 256;
    int c = li >> 6, j = li & 63;
    tile[c][j] = S[c * P_TOT + p0 + j];
  }
  __syncthreads();
#pragma unroll
  for (int i = 0; i < 16; ++i) {
    int li = t + i * 256;
    int j = li >> 6, c = li & 63;
    D[(p0 + j) * CCH + c] = tile[c][j];
  }
}

// ---------------------------------------------------------------------------
// 2) Fuse 1x1 conv1 into deform weights: Wf[(k*64+c)][o] = sum_m w1[o,m]*wd[m,c,k]
// ---------------------------------------------------------------------------
__global__ __launch_bounds__(256) void k_fuse(const float* __restrict__ W1,
                                              const float* __restrict__ Wd,
                                              float* __restrict__ Wf) {
  int idx = blockIdx.x * 256 + threadIdx.x;  // 3136*64 = 200704
  int o = idx & 63;
  int kc = idx >> 6;
  int c = kc & 63, k = kc >> 6;
  float s = 0.f;
#pragma unroll 8
  for (int m = 0; m < 64; ++m) s += W1[o * 64 + m] * Wd[(m * 64 + c) * KDEF + k];
  Wf[idx] = s;
}

// ---------------------------------------------------------------------------
// 3) h_c = irfft2(fft_w[c], s=(8,8)) exactly (numpy convention:
//    ifft over axis0 (e^{+i}), then irfft over axis1 with Hermitian extension)
// ---------------------------------------------------------------------------
__global__ __launch_bounds__(256) void k_h(const float* __restrict__ Fw,
                                           float* __restrict__ Hc) {
  int t = blockIdx.x * 256 + threadIdx.x;  // 64*64 = 4096
  int c = t >> 6, p1 = (t >> 3) & 7, p2 = t & 7;
  const float A8 = 0.78539816339744830962f;  // 2*pi/8
  float acc = 0.f;
  for (int k2 = 0; k2 < 8; ++k2) {
    int k2h = (k2 <= 4) ? k2 : 8 - k2;
    float cs = (k2 <= 4) ? 1.f : -1.f;
    float br = 0.f, bi = 0.f;
    for (int k1 = 0; k1 < 8; ++k1) {
      float wv = Fw[c * 40 + k1 * 5 + k2h];
      float ang = A8 * (float)((k1 * p1) & 7);
      br += wv * cosf(ang);
      bi += wv * sinf(ang);
    }
    bi *= cs;
    float ang2 = A8 * (float)((k2 * p2) & 7);
    acc += br * cosf(ang2) - bi * sinf(ang2);
  }
  Hc[t] = acc * (1.f / 64.f);
}

// ---------------------------------------------------------------------------
// 4) 1x1 proj_in GEMM: Y[p][o] = sum_i A[p][i] * W[o][i]   (WMMA f32)
//    256 threads = 8 waves, 16 pixels/wave, full N=64 per wave.
// ---------------------------------------------------------------------------
__global__ __launch_bounds__(256) void k_gemm64(const float* __restrict__ A,
                                                const float* __restrict__ W,
                                                float* __restrict__ Y) {
  __shared__ float Bs[64 * 64];       // [i][o]
  __shared__ float As[8][16 * LDA];
  int t = threadIdx.x;
  int lane = t & 31, w = t >> 5;
  int p0 = blockIdx.x * 128 + w * 16;
#pragma unroll
  for (int i = 0; i < 16; ++i) {       // W[o][i] -> Bs[i][o], coalesced read
    int li = t + i * 256;
    int o = li >> 6, ii = li & 63;
    Bs[ii * 64 + o] = W[o * 64 + ii];
  }
  float* as = As[w];
#pragma unroll
  for (int i = 0; i < 32; ++i) {
    int li = lane + i * 32;
    int m = li >> 6, c = li & 63;
    as[m * LDA + c] = A[(p0 + m) * CCH + c];
  }
  __syncthreads();
  v8f acc[4] = {};
#pragma unroll
  for (int kk = 0; kk < 16; ++kk) {
    v2f a = afrag(as, lane, kk * 4, LDA);
#pragma unroll
    for (int n = 0; n < 4; ++n)
      acc[n] = wmma4(a, bfrag(Bs, lane, kk * 4, n * 16, 64), acc[n]);
  }
  int col = lane & 15, mb = (lane >> 4) * 8;
#pragma unroll
  for (int n = 0; n < 4; ++n)
#pragma unroll
    for (int r = 0; r < 8; ++r)
      Y[(p0 + mb + r) * CCH + n * 16 + col] = acc[n][r];
}

// ---------------------------------------------------------------------------
// 5) Deformable conv (fused with conv1 via Wf): per k-tap, bilinear-sample a
//    16x64 A-tile (HWC reads are contiguous 256B rows), stream 64x64 Wf rows
//    through LDS, accumulate 16x64 output with WMMA f32. K total = 3136.
// ---------------------------------------------------------------------------
__global__ __launch_bounds__(256) void k_deform(const float* __restrict__ Xh,
                                                const float* __restrict__ Off,
                                                const float* __restrict__ Msk,
                                                const float* __restrict__ Wf,
                                                float* __restrict__ Y) {
  __shared__ float Bs[64 * 64];
  __shared__ float As[8][16 * LDA];
  int t = threadIdx.x, lane = t & 31, w = t >> 5;
  int p0 = blockIdx.x * 128 + w * 16;
  int m = lane >> 1;          // pixel row handled by this lane (2 lanes/pixel)
  int c0 = (lane & 1) * 32;   // channel half
  int p = p0 + m;
  int py = p / WIMG, px = p - py * WIMG;
  float* as = As[w];
  v8f acc[4] = {};
  for (int k = 0; k < KDEF; ++k) {
    __syncthreads();  // protect Bs/As from previous iteration's readers
    const float* wk = Wf + (size_t)(k * 64) * 64;  // Bs[c][o]
#pragma unroll
    for (int i = 0; i < 16; ++i) {
      int li = t + i * 256;
      Bs[li] = wk[li];
    }
    // ---- modulated bilinear sampling for this tap ----
    int ky = k / 7, kx = k - ky * 7;
    float dy = Off[(size_t)(2 * k) * P_TOT + p];
    float dx = Off[(size_t)(2 * k + 1) * P_TOT + p];
    float mk = Msk[(size_t)k * P_TOT + p];
    float yf = (float)(py - 3 + ky) + dy;
    float xf = (float)(px - 3 + kx) + dx;
    float y0f = floorf(yf), x0f = floorf(xf);
    float wy = yf - y0f, wx = xf - x0f;
    int y0 = (int)y0f, x0 = (int)x0f;
    bool vy0 = (y0 >= 0) && (y0 < HIMG);
    bool vy1 = (y0 + 1 >= 0) && (y0 + 1 < HIMG);
    bool vx0 = (x0 >= 0) && (x0 < WIMG);
    bool vx1 = (x0 + 1 >= 0) && (x0 + 1 < WIMG);
    float f00 = (vy0 && vx0) ? (1.f - wy) * (1.f - wx) * mk : 0.f;
    float f01 = (vy0 && vx1) ? (1.f - wy) * wx * mk : 0.f;
    float f10 = (vy1 && vx0) ? wy * (1.f - wx) * mk : 0.f;
    float f11 = (vy1 && vx1) ? wy * wx * mk : 0.f;
    int yc0 = min(max(y0, 0), HIMG - 1), yc1 = min(max(y0 + 1, 0), HIMG - 1);
    int xc0 = min(max(x0, 0), WIMG - 1), xc1 = min(max(x0 + 1, 0), WIMG - 1);
    const float4* q00 = (const float4*)(Xh + ((size_t)(yc0 * WIMG + xc0) * CCH + c0));
    const float4* q01 = (const float4*)(Xh + ((size_t)(yc0 * WIMG + xc1) * CCH + c0));
    const float4* q10 = (const float4*)(Xh + ((size_t)(yc1 * WIMG + xc0) * CCH + c0));
    const float4* q11 = (const float4*)(Xh + ((size_t)(yc1 * WIMG + xc1) * CCH + c0));
    float4* dst = (float4*)(as + m * LDA + c0);
#pragma unroll
    for (int j = 0; j < 8; ++j) {
      float4 a0 = q00[j], a1 = q01[j], a2 = q10[j], a3 = q11[j];
      float4 r;
      r.x = f00 * a0.x + f01 * a1.x + f10 * a2.x + f11 * a3.x;
      r.y = f00 * a0.y + f01 * a1.y + f10 * a2.y + f11 * a3.y;
      r.z = f00 * a0.z + f01 * a1.z + f10 * a2.z + f11 * a3.z;
      r.w = f00 * a0.w + f01 * a1.w + f10 * a2.w + f11 * a3.w;
      dst[j] = r;
    }
    __syncthreads();
#pragma unroll
    for (int kk = 0; kk < 16; ++kk) {
      v2f a = afrag(as, lane, kk * 4, LDA);
#pragma unroll
      for (int n = 0; n < 4; ++n)
        acc[n] = wmma4(a, bfrag(Bs, lane, kk * 4, n * 16, 64), acc[n]);
    }
  }
  int col = lane & 15, mb = (lane >> 4) * 8;
#pragma unroll
  for (int n = 0; n < 4; ++n)
#pragma unroll
    for (int r = 0; r < 8; ++r)
      Y[(size_t)(p0 + mb + r) * CCH + n * 16 + col] = acc[n][r];
}

// ---------------------------------------------------------------------------
// 6) FFT stage as exact 8x8 circular convolution with h_c, fused with +reverse
//    One block per patch; thread = (channel, quarter of patch outputs).
// ---------------------------------------------------------------------------
__global__ __launch_bounds__(256) void k_fft_add(const float* __restrict__ E1,
                                                 const float* __restrict__ Rev,
                                                 const float* __restrict__ Hc,
                                                 float* __restrict__ E) {
  __shared__ float xs[64 * 64];   // [patch-pixel][channel]
  __shared__ float hs[64 * 65];   // [channel][delta], padded
  int t = threadIdx.x;
  int pb = blockIdx.x;            // 24*24 patches
  int pby = pb / 24, pbx = pb - pby * 24;
  int gy0 = pby * 8, gx0 = pbx * 8;
#pragma unroll
  for (int i = 0; i < 16; ++i) {
    int li = t + i * 256;
    int q = li >> 6, c = li & 63;
    int qy = q >> 3, qx = q & 7;
    xs[li] = E1[(size_t)((gy0 + qy) * WIMG + gx0 + qx) * CCH + c];
  }
#pragma unroll
  for (int i = 0; i < 16; ++i) {
    int li = t + i * 256;
    hs[(li >> 6) * 65 + (li & 63)] = Hc[li];
  }
  __syncthreads();
  int c = t & 63, g = t >> 6;
#pragma unroll
  for (int pi = 0; pi < 16; ++pi) {
    int pp = g * 16 + pi;
    int p1 = pp >> 3, p2 = pp & 7;
    float accv = 0.f;
#pragma unroll 8
    for (int q = 0; q < 64; ++q) {
      int q1 = q >> 3, q2 = q & 7;
      int d = ((p1 - q1) & 7) * 8 + ((p2 - q2) & 7);
      accv += xs[q * 64 + c] * hs[c * 65 + d];
    }
    size_t gp = (size_t)((gy0 + p1) * WIMG + gx0 + p2);
    E[gp * CCH + c] = accv + Rev[gp * CCH + c];
  }
}

// ---------------------------------------------------------------------------
// 7) 3x3 conv (pad 1) as 9-tap streamed GEMM: per tap stage shifted 16x64
//    A-tile (zero at image border) + 64x64 weight slice, WMMA accumulate.
// ---------------------------------------------------------------------------
__global__ __launch_bounds__(256) void k_conv3(const float* __restrict__ Xh,
                                               const float* __restrict__ W2,
                                               float* __restrict__ Y) {
  __shared__ float Bs[64 * 64];
  __shared__ float As[8][16 * LDA];
  int t = threadIdx.x, lane = t & 31, w = t >> 5;
  int p0 = blockIdx.x * 128 + w * 16;
  float* as = As[w];
  v8f acc[4] = {};
  for (int tz = 0; tz < 9; ++tz) {
    __syncthreads();
    int ty = tz / 3 - 1, tx = tz % 3 - 1;
#pragma unroll
    for (int i = 0; i < 16; ++i) {   // Bs[c][o] = W2[o][c][tz]
      int li = t + i * 256;
      int o = li >> 6, c = li & 63;
      Bs[c * 64 + o] = W2[(size_t)(o * 64 + c) * 9 + tz];
    }
#pragma unroll
    for (int i = 0; i < 32; ++i) {
      int li = lane + i * 32;
      int m = li >> 6, c = li & 63;
      int pp = p0 + m;
      int pyy = pp / WIMG + ty;
      int pxx = pp - (pp / WIMG) * WIMG + tx;
      bool v = (pyy >= 0) && (pyy < HIMG) && (pxx >= 0) && (pxx < WIMG);
      int pc = min(max(pyy, 0), HIMG - 1) * WIMG + min(max(pxx, 0), WIMG - 1);
      float val = Xh[(size_t)pc * CCH + c];
      as[m * LDA + c] = v ? val : 0.f;
    }
    __syncthreads();
#pragma unroll
    for (int kk = 0; kk < 16; ++kk) {
      v2f a = afrag(as, lane, kk * 4, LDA);
#pragma unroll
      for (int n = 0; n < 4; ++n)
        acc[n] = wmma4(a, bfrag(Bs, lane, kk * 4, n * 16, 64), acc[n]);
    }
  }
  int col = lane & 15, mb = (lane >> 4) * 8;
#pragma unroll
  for (int n = 0; n < 4; ++n)
#pragma unroll
    for (int r = 0; r < 8; ++r)
      Y[(size_t)(p0 + mb + r) * CCH + n * 16 + col] = acc[n][r];
}

// ---------------------------------------------------------------------------
// 8) proj_out (K=128, N=128) + SimpleGate fused; output transposed to NCHW
//    through LDS. 128 threads = 4 waves to keep static LDS < 64KB.
// ---------------------------------------------------------------------------
__global__ __launch_bounds__(128) void k_out(const float* __restrict__ E2,
                                             const float* __restrict__ Dh,
                                             const float* __restrict__ Wo,
                                             float* __restrict__ Out) {
  __shared__ float Bs[128 * 16];     // one 128x16 N-slice at a time
  __shared__ float As[4][16 * LDA2];
  int t = threadIdx.x, lane = t & 31, w = t >> 5;
  int p0 = blockIdx.x * 64 + w * 16;
  float* as = As[w];
#pragma unroll
  for (int i = 0; i < 64; ++i) {     // A = [e2 | dec], 16x128
    int li = lane + i * 32;
    int m = li >> 7, c = li & 127;
    float v = (c < 64) ? E2[(size_t)(p0 + m) * CCH + c]
                       : Dh[(size_t)(p0 + m) * CCH + (c - 64)];
    as[m * LDA2 + c] = v;
  }
  v8f acc[8] = {};
  for (int n = 0; n < 8; ++n) {
    __syncthreads();
#pragma unroll
    for (int i = 0; i < 16; ++i) {   // Bs[i][oo] = Wo[n*16+oo][i]
      int li = t + i * 128;
      int oo = li >> 7, ii = li & 127;
      Bs[ii * 16 + oo] = Wo[(size_t)(n * 16 + oo) * 128 + ii];
    }
    __syncthreads();
#pragma unroll
    for (int kk = 0; kk < 32; ++kk)
      acc[n] = wmma4(afrag(as, lane, kk * 4, LDA2),
                     bfrag(Bs, lane, kk * 4, 0, 16), acc[n]);
  }
  __syncthreads();
  // SimpleGate: out[o] = f[o] * f[o+64]; o and o+64 share lane in acc[n]/acc[n+4]
  int col = lane & 15, mb = (lane >> 4) * 8;
#pragma unroll
  for (int n = 0; n < 4; ++n)
#pragma unroll
    for (int r = 0; r < 8; ++r)
      as[(mb + r) * LDA2 + n * 16 + col] = acc[n][r] * acc[n + 4][r];
  __syncthreads();
#pragma unroll
  for (int i = 0; i < 32; ++i) {     // 16 px x 64 ch, transposed NCHW store
    int li = lane + i * 32;
    int m = li & 15, o = li >> 4;
    Out[(size_t)o * P_TOT + p0 + m] = as[m * LDA2 + o];
  }
}

// ---------------------------------------------------------------------------
// Host launcher. Workspace layout (floats): 6 HWC tensors + fused W + h.
// Total = 6*2359296 + 200704 + 4096 floats = ~57.5 MB.
// ---------------------------------------------------------------------------
extern "C" void kernel_launch(void* const* d_in, const int* in_sizes, int n_in,
                              void* d_out, int out_size, void* d_ws,
                              size_t ws_size, hipStream_t stream) {
  const float* enc  = (const float*)d_in[0];
  const float* dec  = (const float*)d_in[1];
  const float* ioff = (const float*)d_in[2];
  const float* iwt  = (const float*)d_in[3];
  const float* w_in = (const float*)d_in[4];
  const float* fftw = (const float*)d_in[5];
  const float* w_o  = (const float*)d_in[6];
  const float* w_df = (const float*)d_in[7];
  const float* w_c1 = (const float*)d_in[8];
  const float* w_c2 = (const float*)d_in[9];
  float* out = (float*)d_out;

  size_t PC = (size_t)P_TOT * CCH;
  float* ws    = (float*)d_ws;
  float* enc_h = ws;
  float* dec_h = enc_h + PC;
  float* e1    = dec_h + PC;
  float* rev   = e1 + PC;
  float* e     = rev + PC;
  float* e2    = e + PC;
  float* Wf    = e2 + PC;            // 3136*64
  float* Hc    = Wf + 3136 * 64;     // 64*64

  k_transpose<<<P_TOT / 64, 256, 0, stream>>>(enc, enc_h);
  k_transpose<<<P_TOT / 64, 256, 0, stream>>>(dec, dec_h);
  k_fuse<<<(3136 * 64) / 256, 256, 0, stream>>>(w_c1, w_df, Wf);
  k_h<<<16, 256, 0, stream>>>(fftw, Hc);
  k_gemm64<<<P_TOT / 128, 256, 0, stream>>>(enc_h, w_in, e1);
  k_deform<<<P_TOT / 128, 256, 0, stream>>>(enc_h, ioff, iwt, Wf, rev);
  k_fft_add<<<576, 256, 0, stream>>>(e1, rev, Hc, e);
  k_conv3<<<P_TOT / 128, 256, 0, stream>>>(e, w_c2, e2);
  k_out<<<P_TOT / 64, 128, 0, stream>>>(e2, dec_h, w_o, out);
}